// S4Block_75058848465497
// MI455X (gfx1250) — compile-verified
//
#include <hip/hip_runtime.h>
#include <math.h>

#define BATCH   8
#define SEQ     2048
#define DM      1024
#define NS      64          // d_state
#define LDP     66          // padded LDS stride for 64x64 matrices (expm)

typedef float v2f __attribute__((ext_vector_type(2)));
typedef float v4f __attribute__((ext_vector_type(4)));
typedef float v8f __attribute__((ext_vector_type(8)));

__device__ __forceinline__ v8f wmma4(v2f a, v2f b, v8f c) {
    // D = A(16x4 f32) x B(4x16 f32) + C(16x16 f32)
    return __builtin_amdgcn_wmma_f32_16x16x4_f32(
        /*neg_a=*/false, a, /*neg_b=*/false, b,
        /*c_mod=*/(short)0, c, /*reuse_a=*/false, /*reuse_b=*/false);
}

__device__ __forceinline__ float fast_tanh(float x) {
#if __has_builtin(__builtin_amdgcn_tanhf)
    return __builtin_amdgcn_tanhf(x);
#else
    float r;
    // gfx1250 TRANS op; v_nop covers the trans-result hazard window.
    asm volatile("v_tanh_f32 %0, %1\n\tv_nop" : "=v"(r) : "v"(x));
    return r;
#endif
}

// ---------------------------------------------------------------------------
// Kernel 1: delta = softplus(log_delta); delta_mean; ||A||_inf; zero A_bar acc
// ---------------------------------------------------------------------------
__global__ __launch_bounds__(1024)
void prep0_kernel(const float* __restrict__ log_delta, const float* __restrict__ A,
                  float* __restrict__ delta, float* __restrict__ dmean,
                  float* __restrict__ anorm, float* __restrict__ Abar_acc) {
    __shared__ float red[1024];
    int tid = threadIdx.x;
    float ld = log_delta[tid];
    float d  = (ld > 20.0f) ? ld : log1pf(expf(ld));   // stable softplus
    delta[tid] = d;
    red[tid] = d;
    __syncthreads();
    for (int off = 512; off > 0; off >>= 1) {
        if (tid < off) red[tid] += red[tid + off];
        __syncthreads();
    }
    if (tid == 0) dmean[0] = red[0] * (1.0f / (float)DM);
    __syncthreads();
    if (tid < NS) {
        float s = 0.0f;
        for (int j = 0; j < NS; ++j) s += fabsf(A[tid * NS + j]);
        red[tid] = s;
    }
    __syncthreads();
    if (tid == 0) {
        float m = 0.0f;
        for (int r = 0; r < NS; ++r) m = fmaxf(m, red[r]);
        anorm[0] = m;
    }
    for (int e = tid; e < NS * NS; e += 1024) Abar_acc[e] = 0.0f;
}

// ---------------------------------------------------------------------------
// Kernel 2: per-channel expm(delta_i * A), scaling-and-squaring + Taylor-12.
// ---------------------------------------------------------------------------
__device__ __forceinline__
void mm64(const float* __restrict__ M1, const float* __restrict__ M2,
          float* __restrict__ Mo, int tid) {
    int r  = tid >> 2;
    int c0 = (tid & 3) << 4;
    float acc[16];
#pragma unroll
    for (int j = 0; j < 16; ++j) acc[j] = 0.0f;
    for (int k = 0; k < NS; ++k) {
        float a = M1[r * LDP + k];
#pragma unroll
        for (int j = 0; j < 16; ++j)
            acc[j] = fmaf(a, M2[k * LDP + c0 + j], acc[j]);
    }
#pragma unroll
    for (int j = 0; j < 16; ++j) Mo[r * LDP + c0 + j] = acc[j];
}

__global__ __launch_bounds__(256)
void expm_accum_kernel(const float* __restrict__ A, const float* __restrict__ delta,
                       const float* __restrict__ anorm_p, float* __restrict__ Abar_acc) {
    __shared__ float T[NS * LDP], P[NS * LDP], W[NS * LDP];
    const int tid = threadIdx.x;
    const float dt  = delta[blockIdx.x];
    const float nrm = dt * anorm_p[0];
    int s = 0;
    if (nrm > 0.25f) s = (int)ceilf(log2f(nrm * 4.0f));
    const float scale = dt * exp2f(-(float)s);

    for (int e = tid; e < NS * NS; e += 256) {
        int r = e >> 6, c = e & 63;
        T[r * LDP + c] = A[e] * scale;
    }
    __syncthreads();

    const int KMAX = 12;
    for (int e = tid; e < NS * NS; e += 256) {
        int r = e >> 6, c = e & 63;
        P[r * LDP + c] = ((r == c) ? 1.0f : 0.0f) + T[r * LDP + c] * (1.0f / (float)KMAX);
    }
    __syncthreads();
    for (int k = KMAX - 1; k >= 1; --k) {      // Horner: P = I + (T@P)/k
        mm64(T, P, W, tid);
        __syncthreads();
        float inv = 1.0f / (float)k;
        for (int e = tid; e < NS * NS; e += 256) {
            int r = e >> 6, c = e & 63;
            P[r * LDP + c] = ((r == c) ? 1.0f : 0.0f) + W[r * LDP + c] * inv;
        }
        __syncthreads();
    }
    for (int q = 0; q < s; ++q) {              // repeated squaring
        mm64(P, P, W, tid);
        __syncthreads();
        for (int e = tid; e < NS * NS; e += 256) {
            int r = e >> 6, c = e & 63;
            P[r * LDP + c] = W[r * LDP + c];
        }
        __syncthreads();
    }
    for (int e = tid; e < NS * NS; e += 256) {
        int r = e >> 6, c = e & 63;
        atomicAdd(&Abar_acc[e], P[r * LDP + c] * (1.0f / (float)DM));
    }
}

// ---------------------------------------------------------------------------
// Kernel 3: pair-interleaved B-operand tables so each lane's two k-values are
// adjacent -> single b64 fragment loads, lane-contiguous (coalesced).
//   AbP[p*128 + 2n+i] = A_bar[n][2p+i]          (p<32)   B-op of h @ A_barT
//   BbP[p*128 + 2n+i] = dmean * B[n][2p+i]      (p<512)  B-op of x @ B_barT
//   CP [p*2048 + 2d+i] = C[d][2p+i]             (p<32)   B-op of h @ C^T
// ---------------------------------------------------------------------------
__global__ void prep1_kernel(const float* __restrict__ Abar, const float* __restrict__ B,
                             const float* __restrict__ C, const float* __restrict__ dmean,
                             float* __restrict__ AbP, float* __restrict__ BbP,
                             float* __restrict__ CP) {
    int idx  = blockIdx.x * blockDim.x + threadIdx.x;
    int nthr = gridDim.x * blockDim.x;
    float dm = dmean[0];
    for (int e = idx; e < NS * NS; e += nthr) {
        int p = e >> 7, rem = e & 127, n = rem >> 1, i = rem & 1;
        AbP[e] = Abar[n * NS + 2 * p + i];
    }
    for (int e = idx; e < DM * NS; e += nthr) {
        int p = e >> 7, rem = e & 127, n = rem >> 1, i = rem & 1;
        BbP[e] = dm * B[n * DM + 2 * p + i];
    }
    for (int e = idx; e < NS * DM; e += nthr) {
        int p = e >> 11, rem = e & 2047, d = rem >> 1, i = rem & 1;
        CP[e] = C[d * NS + 2 * p + i];
    }
}

// ---------------------------------------------------------------------------
// Kernel 4: xbF = (x @ B_barT) in D-fragment layout [t][wave][lane][v].
// One block per s. Row-8 of the LDS stage is a zero row: pad lanes read it
// unconditionally (address select, no EXEC masking).
// ---------------------------------------------------------------------------
#define XS_LD 1028
__global__ __launch_bounds__(128)
void bproj_kernel(const float* __restrict__ x, const float* __restrict__ BbP,
                  float* __restrict__ xbF) {
    __shared__ float xs[(BATCH + 1) * XS_LD];
    const int s    = blockIdx.x;
    const int tid  = threadIdx.x;
    const int lane = tid & 31;
    const int half = lane >> 4;
    const int lr   = lane & 15;
    const int wave = tid >> 5;
    const int n0   = wave << 4;

    // stage x rows (vectorized b128), zero pad row 8
    for (int e4 = tid; e4 < (BATCH * DM) / 4; e4 += 128) {
        int e = e4 << 2;
        int b = e >> 10, d = e & 1023;
        *(v4f*)&xs[b * XS_LD + d] = *(const v4f*)&x[((size_t)b * SEQ + s) * DM + d];
    }
    for (int e4 = tid; e4 < DM / 4; e4 += 128)
        *(v4f*)&xs[BATCH * XS_LD + (e4 << 2)] = (v4f){0.f, 0.f, 0.f, 0.f};
    __syncthreads();

    const int row = (lr < BATCH) ? lr : BATCH;       // pad lanes -> zero row
    const float* xrow = &xs[row * XS_LD + 2 * half];
    const float* bcol = BbP + 2 * (n0 + lr) + half * 128;

    v8f acc0 = {0.f, 0.f, 0.f, 0.f, 0.f, 0.f, 0.f, 0.f};
    v8f acc1 = acc0;
    for (int k = 0; k < DM; k += 8) {                // two k-steps per iter
        v2f a0 = *(const v2f*)(xrow + k);
        v2f b0 = *(const v2f*)(bcol + (k >> 1) * 128);
        v2f a1 = *(const v2f*)(xrow + k + 4);
        v2f b1 = *(const v2f*)(bcol + ((k >> 1) + 2) * 128);
        acc0 = wmma4(a0, b0, acc0);
        acc1 = wmma4(a1, b1, acc1);
    }
    v8f acc = acc0 + acc1;
    *(v8f*)&xbF[(size_t)s * 1024 + wave * 256 + lane * 8] = acc;
}

// ---------------------------------------------------------------------------
// Kernel 5: sequential scan. Single workgroup, 4 waves, h in LDS (stride 68).
// A_barT fragments in registers; per-step xb prefetched one step ahead as v8f;
// 16 WMMAs split across two accumulator chains; hardware tanh.
// ---------------------------------------------------------------------------
#define H_LD 68
__global__ __launch_bounds__(128)
void scan_kernel(const float* __restrict__ AbP, const float* __restrict__ xbF,
                 float* __restrict__ HtF) {
    __shared__ float h[16 * H_LD];
    const int tid  = threadIdx.x;
    const int lane = tid & 31;
    const int half = lane >> 4;
    const int lr   = lane & 15;
    const int wave = tid >> 5;
    const int n0   = wave << 4;
    const int fofs = wave * 256 + lane * 8;          // fragment offset in [t] row

    for (int e = tid; e < 16 * H_LD; e += 128) h[e] = 0.0f;

    v2f bf[16];                                      // A_barT fragments (resident)
#pragma unroll
    for (int ks = 0; ks < 16; ++ks)
        bf[ks] = *(const v2f*)&AbP[(2 * ks + half) * 128 + 2 * (n0 + lr)];

    const float* hrow = &h[lr * H_LD + 2 * half];
    __syncthreads();

    v8f xv = *(const v8f*)&xbF[fofs];                // step 0 input
    for (int t = 0; t < SEQ; ++t) {
        v8f xnext = *(const v8f*)&xbF[(size_t)(t + 1) * 1024 + fofs]; // slack row at t=SEQ
        v8f acc0 = {0.f, 0.f, 0.f, 0.f, 0.f, 0.f, 0.f, 0.f};
        v8f acc1 = acc0;
#pragma unroll
        for (int ks = 0; ks < 16; ks += 2) {
            v2f a0 = *(const v2f*)(hrow + 4 * ks);
            v2f a1 = *(const v2f*)(hrow + 4 * ks + 4);
            acc0 = wmma4(a0, bf[ks],     acc0);
            acc1 = wmma4(a1, bf[ks + 1], acc1);
        }
        v8f acc = acc0 + acc1;
        __syncthreads();                             // all h reads complete
        v8f hv;
#pragma unroll
        for (int v = 0; v < 8; ++v) {
            float val = fast_tanh(acc[v] + xv[v]);
            hv[v] = val;
            h[(v + 8 * half) * H_LD + n0 + lr] = val;
        }
        *(v8f*)&HtF[(size_t)t * 1024 + fofs] = hv;   // 2x b128 store
        xv = xnext;
        __syncthreads();                             // h fully updated
    }
}

// ---------------------------------------------------------------------------
// Kernel 6: Y[b][s][d] = (H_t[s] @ C^T)[b][d] + x[b][s][d] * D[d]
// One block per s; 4 waves x 16 D-tiles each.
// ---------------------------------------------------------------------------
__global__ __launch_bounds__(128)
void readout_kernel(const float* __restrict__ HtF, const float* __restrict__ CP,
                    const float* __restrict__ x, const float* __restrict__ Dv,
                    float* __restrict__ y) {
    __shared__ float hs[16 * H_LD];
    const int s    = blockIdx.x;
    const int tid  = threadIdx.x;
    const int lane = tid & 31;
    const int half = lane >> 4;
    const int lr   = lane & 15;
    const int wave = tid >> 5;

    // de-fragment H_t[s] into (m, n) layout in LDS (coalesced global reads)
    for (int e = tid; e < 16 * NS; e += 128) {
        int w = e >> 8, l = (e >> 3) & 31, v = e & 7;
        int m = v + 8 * (l >> 4);
        int n = (w << 4) + (l & 15);
        hs[m * H_LD + n] = HtF[(size_t)s * 1024 + e];
    }
    __syncthreads();

    const float* hrow = &hs[lr * H_LD + 2 * half];
    for (int j = 0; j < 16; ++j) {
        const int d0 = (wave + 4 * j) << 4;
        const float* ccol = CP + half * 2048 + 2 * (d0 + lr);
        v8f acc0 = {0.f, 0.f, 0.f, 0.f, 0.f, 0.f, 0.f, 0.f};
        v8f acc1 = acc0;
#pragma unroll
        for (int ks = 0; ks < 16; ks += 2) {
            v2f a0 = *(const v2f*)(hrow + 4 * ks);
            v2f a1 = *(const v2f*)(hrow + 4 * ks + 4);
            v2f b0 = *(const v2f*)(ccol + (2 * ks) * 2048);
            v2f b1 = *(const v2f*)(ccol + (2 * ks + 2) * 2048);
            acc0 = wmma4(a0, b0, acc0);
            acc1 = wmma4(a1, b1, acc1);
        }
        v8f acc = acc0 + acc1;
#pragma unroll
        for (int v = 0; v < 8; ++v) {
            int m = v + 8 * half;
            if (m < BATCH) {
                int d = d0 + lr;
                size_t xi = ((size_t)m * SEQ + s) * DM + d;
                y[xi] = acc[v] + x[xi] * Dv[d];
            }
        }
    }
}

// ---------------------------------------------------------------------------
// Host orchestration
// ---------------------------------------------------------------------------
extern "C" void kernel_launch(void* const* d_in, const int* in_sizes, int n_in,
                              void* d_out, int out_size, void* d_ws, size_t ws_size,
                              hipStream_t stream) {
    const float* x   = (const float*)d_in[0];   // (8,2048,1024)
    const float* A   = (const float*)d_in[1];   // (64,64)
    const float* B   = (const float*)d_in[2];   // (64,1024)
    const float* C   = (const float*)d_in[3];   // (1024,64)
    const float* Dv  = (const float*)d_in[4];   // (1024,)
    const float* ldl = (const float*)d_in[5];   // (1024,)
    float* y = (float*)d_out;

    float* ws      = (float*)d_ws;
    float* w_delta = ws;                        // 1024
    float* w_dmean = ws + 1024;                 // 1
    float* w_anorm = ws + 1025;                 // 1
    float* w_Abar  = ws + 1088;                 // 4096
    float* w_AbP   = ws + 5184;                 // 4096
    float* w_BbP   = ws + 9280;                 // 65536
    float* w_CP    = ws + 74816;                // 65536
    float* w_xbF   = ws + 140352;               // (2048+1)*1024 = 2098176 (1 slack row)
    float* w_HtF   = ws + 2238528;              // 2048*1024 = 2097152
    // total ~17.3 MB of workspace

    prep0_kernel<<<1, 1024, 0, stream>>>(ldl, A, w_delta, w_dmean, w_anorm, w_Abar);
    expm_accum_kernel<<<DM, 256, 0, stream>>>(A, w_delta, w_anorm, w_Abar);
    prep1_kernel<<<64, 256, 0, stream>>>(w_Abar, B, C, w_dmean, w_AbP, w_BbP, w_CP);
    bproj_kernel<<<SEQ, 128, 0, stream>>>(x, w_BbP, w_xbF);
    scan_kernel<<<1, 128, 0, stream>>>(w_AbP, w_xbF, w_HtF);
    readout_kernel<<<SEQ, 128, 0, stream>>>(w_HtF, w_CP, x, Dv, y);

    (void)in_sizes; (void)n_in; (void)out_size; (void)ws_size;
}